// RossiDirSAGEConv_83408264888595
// MI455X (gfx1250) — compile-verified
//
#include <hip/hip_runtime.h>

typedef float v2f __attribute__((ext_vector_type(2)));
typedef float v8f __attribute__((ext_vector_type(8)));

#define N_NODES 50000
#define N_EDGES 800000
#define D_IN    128
#define D_OUT   128
#define K_TOT   384
#define LDH     388   // padded LDS row stride (388 mod 64 = 4 -> 16 rows hit distinct banks)

// ---------------------------------------------------------------------------
// Kernel 1: zero the accumulator workspace (2*N*128 sums + 2*N degrees)
// ---------------------------------------------------------------------------
__global__ void zero_ws_kernel(float4* __restrict__ ws, long n4) {
    long i      = (long)blockIdx.x * blockDim.x + threadIdx.x;
    long stride = (long)gridDim.x * blockDim.x;
    float4 z = make_float4(0.f, 0.f, 0.f, 0.f);
    for (; i < n4; i += stride) ws[i] = z;
}

// ---------------------------------------------------------------------------
// Kernel 2: edge scatter. One wave32 per edge; each lane owns 4 consecutive
// features. Atomic f32 adds land in L2 (accumulators are 51MB << 192MB L2).
// ---------------------------------------------------------------------------
__global__ void __launch_bounds__(256) edge_scatter_kernel(
    const float* __restrict__ feat,
    const int*   __restrict__ src,
    const int*   __restrict__ dst,
    float* __restrict__ fwd_sum,
    float* __restrict__ bwd_sum,
    float* __restrict__ fwd_deg,
    float* __restrict__ bwd_deg) {
    const int lane = threadIdx.x & 31;
    const int edge = blockIdx.x * (blockDim.x >> 5) + (threadIdx.x >> 5);
    if (edge >= N_EDGES) return;

    const int s = src[edge];
    const int d = dst[edge];
    const int f = lane * 4;  // 32 lanes x 4 floats = 128 features

    // forward: message feat[src] accumulated at dst
    float4 v = *reinterpret_cast<const float4*>(&feat[(long)s * D_IN + f]);
    float* p = &fwd_sum[(long)d * D_IN + f];
    atomicAdd(p + 0, v.x); atomicAdd(p + 1, v.y);
    atomicAdd(p + 2, v.z); atomicAdd(p + 3, v.w);

    // backward (reversed graph): message feat[dst] accumulated at src
    float4 u = *reinterpret_cast<const float4*>(&feat[(long)d * D_IN + f]);
    float* q = &bwd_sum[(long)s * D_IN + f];
    atomicAdd(q + 0, u.x); atomicAdd(q + 1, u.y);
    atomicAdd(q + 2, u.z); atomicAdd(q + 3, u.w);

    if (lane == 0) {
        atomicAdd(&fwd_deg[d], 1.0f);
        atomicAdd(&bwd_deg[s], 1.0f);
    }
}

// ---------------------------------------------------------------------------
// Kernel 3: fused normalize + concat + GEMM via V_WMMA_F32_16X16X4_F32.
// Block = 256 threads = 8 waves. Tile = 16 nodes x 128 outputs.
// Wave w computes the 16x16 output tile for columns [16w, 16w+16).
// ---------------------------------------------------------------------------
__global__ void __launch_bounds__(256) sage_wmma_kernel(
    const float* __restrict__ feat,
    const float* __restrict__ W,      // [128, 384] row-major
    const float* __restrict__ bias,   // [128]
    const float* __restrict__ fwd_sum,
    const float* __restrict__ bwd_sum,
    const float* __restrict__ fwd_deg,
    const float* __restrict__ bwd_deg,
    float* __restrict__ out) {        // [N, 128]
    __shared__ float lh[16 * LDH];    // 16 x 384 h-tile, padded rows

    const int tid   = threadIdx.x;
    const int node0 = blockIdx.x * 16;

    // Stage h = [feat | fwd_sum/deg | bwd_sum/deg] for 16 nodes, coalesced.
    for (int idx = tid; idx < 16 * K_TOT; idx += 256) {
        const int r    = idx / K_TOT;
        const int k    = idx - r * K_TOT;
        const int node = node0 + r;
        float v;
        if (k < D_IN) {
            v = feat[(long)node * D_IN + k];
        } else if (k < 2 * D_IN) {
            const float dg = fwd_deg[node];
            v = (dg > 0.f) ? fwd_sum[(long)node * D_IN + (k - D_IN)] / dg : 0.f;
        } else {
            const float dg = bwd_deg[node];
            v = (dg > 0.f) ? bwd_sum[(long)node * D_IN + (k - 2 * D_IN)] / dg : 0.f;
        }
        lh[r * LDH + k] = v;
    }
    __syncthreads();

    const int lane  = tid & 31;
    const int wave  = tid >> 5;             // 8 column tiles of 16
    const int row   = lane & 15;            // A-matrix M index (node in tile)
    const int col   = (lane & 15) + wave * 16;  // B/D N index (output channel)
    const int khalf = (lane >> 4) * 2;      // lanes 16-31 carry K+2, K+3

    // C seeded with the bias: every C/D VGPR of a lane maps to the same N.
    v8f c;
    const float bb = bias[col];
#pragma unroll
    for (int i = 0; i < 8; ++i) c[i] = bb;

    const float* wrow = &W[(long)col * K_TOT];

#pragma unroll 4
    for (int k0 = 0; k0 < K_TOT; k0 += 4) {
        v2f a, b;
        // A 16x4 f32 layout: lane row, VGPR0=K(khalf), VGPR1=K(khalf+1)
        a.x = lh[row * LDH + k0 + khalf];
        a.y = lh[row * LDH + k0 + khalf + 1];
        // B 4x16: B[k][n] = W[n][k]; aligned 8B float2 load, L2/WGP$ resident
        const float2 wv = *reinterpret_cast<const float2*>(&wrow[k0 + khalf]);
        b.x = wv.x;
        b.y = wv.y;
        // (neg_a, A, neg_b, B, c_mod, C, reuse_a, reuse_b)
        c = __builtin_amdgcn_wmma_f32_16x16x4_f32(
                false, a, false, b, (short)0, c, false, false);
    }

    // D layout: VGPR r -> M = r (lanes 0-15) or M = 8+r (lanes 16-31), N = col
    const int mbase = (lane >> 4) * 8;
#pragma unroll
    for (int r = 0; r < 8; ++r) {
        const int node = node0 + mbase + r;
        out[(long)node * D_OUT + col] = c[r];
    }
}

// ---------------------------------------------------------------------------
extern "C" void kernel_launch(void* const* d_in, const int* in_sizes, int n_in,
                              void* d_out, int out_size, void* d_ws, size_t ws_size,
                              hipStream_t stream) {
    const float* feat = (const float*)d_in[0];
    const int*   src  = (const int*)  d_in[1];
    const int*   dst  = (const int*)  d_in[2];
    const float* W    = (const float*)d_in[3];
    const float* b    = (const float*)d_in[4];
    float*       out  = (float*)d_out;

    // Workspace layout: fwd_sum[N*128] | bwd_sum[N*128] | fwd_deg[N] | bwd_deg[N]
    float* fwd_sum = (float*)d_ws;
    float* bwd_sum = fwd_sum + (long)N_NODES * D_IN;
    float* fwd_deg = bwd_sum + (long)N_NODES * D_IN;
    float* bwd_deg = fwd_deg + N_NODES;

    const long total_f = 2L * N_NODES * D_IN + 2L * N_NODES;  // 12.9M floats
    zero_ws_kernel<<<2048, 256, 0, stream>>>((float4*)d_ws, total_f / 4);

    edge_scatter_kernel<<<(N_EDGES + 7) / 8, 256, 0, stream>>>(
        feat, src, dst, fwd_sum, bwd_sum, fwd_deg, bwd_deg);

    sage_wmma_kernel<<<N_NODES / 16, 256, 0, stream>>>(
        feat, W, b, fwd_sum, bwd_sum, fwd_deg, bwd_deg, out);
}